// Net_48335561949837
// MI455X (gfx1250) — compile-verified
//
#include <hip/hip_runtime.h>

// fp32 WMMA operand types for V_WMMA_F32_16X16X4_F32
typedef float v2f __attribute__((ext_vector_type(2)));
typedef float v8f __attribute__((ext_vector_type(8)));

#define STRIDE 141   // LDS row stride (>= 140, odd to spread banks)
#define NWAVES 8     // 256 threads / wave32

// One conv3x3 stage computed as banded-B WMMA GEMMs:
//   dst[r,c] = sum_{dy,dx} src[r+dy, c+dx] * w[dy][dx]
// For each 16x16 output tile and each dy: Y += A(16x18) * B(18x16),
// A[i,k] = src[r0+dy+i, c0+k], B[k,j] = w[dy][k-j] (band 0..2), K split 5x4.
__device__ __forceinline__ void conv3x3_wmma_stage(
    const float* __restrict__ src, float* __restrict__ dst,
    int Sout, int Sin, int gy0, int gx0, int limit,
    const float* __restrict__ w, int wave, int lane)
{
    float wk[3][3];
#pragma unroll
    for (int dy = 0; dy < 3; ++dy)
#pragma unroll
        for (int dx = 0; dx < 3; ++dx)
            wk[dy][dx] = w[dy * 3 + dx];

    const int j    = lane & 15;   // N index (B/D), also M index for A rows
    const int half = lane >> 4;   // K/M half-select

    // Precompute banded B operands: per (dy, K-chunk) two VGPRs.
    // B VGPR0 holds rows K = kb + 2*half, VGPR1 holds K = kb + 2*half + 1.
    float b0v[3][5], b1v[3][5];
#pragma unroll
    for (int dy = 0; dy < 3; ++dy) {
#pragma unroll
        for (int ci = 0; ci < 5; ++ci) {
            int k0 = ci * 4 + 2 * half;
            int d0 = k0 - j;
            int d1 = d0 + 1;
            b0v[dy][ci] = (d0 >= 0 && d0 < 3) ? wk[dy][d0] : 0.0f;
            b1v[dy][ci] = (d1 >= 0 && d1 < 3) ? wk[dy][d1] : 0.0f;
        }
    }

    const int nT = (Sout + 15) >> 4;          // tiles per edge (last tile clamped)
    for (int t = wave; t < nT * nT; t += NWAVES) {
        int r0 = (t / nT) * 16; if (r0 > Sout - 16) r0 = Sout - 16;
        int c0 = (t % nT) * 16; if (c0 > Sout - 16) c0 = Sout - 16;

        v8f acc = {0.f, 0.f, 0.f, 0.f, 0.f, 0.f, 0.f, 0.f};
#pragma unroll
        for (int dy = 0; dy < 3; ++dy) {
            const int rowBase = (r0 + dy + j) * STRIDE;  // A row: M = lane&15
#pragma unroll
            for (int ci = 0; ci < 5; ++ci) {
                int cb = c0 + ci * 4 + 2 * half;
                // clamp columns (band weight is 0 there; avoid NaN*0 from
                // uninitialized LDS past the valid width)
                int cA = (cb     < Sin - 1) ? cb     : (Sin - 1);
                int cB = (cb + 1 < Sin - 1) ? cb + 1 : (Sin - 1);
                v2f a;
                a.x = src[rowBase + cA];
                a.y = src[rowBase + cB];
                v2f b;
                b.x = b0v[dy][ci];
                b.y = b1v[dy][ci];
                acc = __builtin_amdgcn_wmma_f32_16x16x4_f32(
                    false, a, false, b, (short)0, acc, false, false);
            }
        }

        // Store D (M = v + 8*half, N = lane&15), zeroing out-of-image halo
        // so the next stage sees the conv's implicit zero padding.
#pragma unroll
        for (int v = 0; v < 8; ++v) {
            int mr = v + 8 * half;
            int gy = gy0 + r0 + mr;
            int gx = gx0 + c0 + j;
            float val = (gy >= 0 && gy < limit && gx >= 0 && gx < limit)
                        ? acc[v] : 0.0f;
            dst[(r0 + mr) * STRIDE + (c0 + j)] = val;
        }
    }
}

// Fused: conv(w1) -> conv(w2) -> maxpool2 -> conv(w3) -> conv(w4) -> maxpool2.
// One workgroup per 32x32 final-output tile (grid 4*32*32 = 4096 WGs).
__global__ __launch_bounds__(256) void fused_conv_pool(
    const float* __restrict__ x,
    const float* __restrict__ w1, const float* __restrict__ w2,
    const float* __restrict__ w3, const float* __restrict__ w4,
    float* __restrict__ out)
{
    __shared__ float buf0[140 * STRIDE];  // X -> B -> C
    __shared__ float buf1[138 * STRIDE];  // A -> P -> D

    const int b    = blockIdx.x;
    const int n    = b >> 10;
    const int ty   = (b >> 5) & 31;
    const int tx   = b & 31;
    const int tid  = threadIdx.x;
    const int wave = tid >> 5;
    const int lane = tid & 31;

    // Stage 0: load 140x140 input halo tile, zero-filled outside the image.
    {
        const int oy = ty * 128 - 6, ox = tx * 128 - 6;
        const long imgBase = (long)n * 4096 * 4096;
        for (int i = tid; i < 140 * 140; i += 256) {
            int r = i / 140, c = i - r * 140;
            int gy = oy + r, gx = ox + c;
            float v = 0.0f;
            if ((unsigned)gy < 4096u && (unsigned)gx < 4096u)
                v = x[imgBase + (long)gy * 4096 + gx];
            buf0[r * STRIDE + c] = v;
        }
    }
    __syncthreads();

    // conv1: X(140, origin ty*128-6) -> A(138, origin ty*128-5)
    conv3x3_wmma_stage(buf0, buf1, 138, 140, ty * 128 - 5, tx * 128 - 5, 4096,
                       w1, wave, lane);
    __syncthreads();

    // conv2: A(138) -> B(136, origin ty*128-4)
    conv3x3_wmma_stage(buf1, buf0, 136, 138, ty * 128 - 4, tx * 128 - 4, 4096,
                       w2, wave, lane);
    __syncthreads();

    // pool1: B(136, full res) -> P(68, half res, origin ty*64-2)
    {
        const int hy0 = ty * 64 - 2, hx0 = tx * 64 - 2;
        for (int i = tid; i < 68 * 68; i += 256) {
            int py = i / 68, px = i - py * 68;
            float v = 0.0f;
            if ((unsigned)(hy0 + py) < 2048u && (unsigned)(hx0 + px) < 2048u) {
                const float* p = &buf0[(2 * py) * STRIDE + 2 * px];
                v = fmaxf(fmaxf(p[0], p[1]), fmaxf(p[STRIDE], p[STRIDE + 1]));
            }
            buf1[py * STRIDE + px] = v;
        }
    }
    __syncthreads();

    // conv3: P(68) -> C(66, origin ty*64-1), half-res image is 2048^2
    conv3x3_wmma_stage(buf1, buf0, 66, 68, ty * 64 - 1, tx * 64 - 1, 2048,
                       w3, wave, lane);
    __syncthreads();

    // conv4: C(66) -> D(64, origin ty*64)
    conv3x3_wmma_stage(buf0, buf1, 64, 66, ty * 64, tx * 64, 2048,
                       w4, wave, lane);
    __syncthreads();

    // pool2: D(64) -> final 32x32 tile of out (quarter res, 1024^2)
    {
        const long outBase = ((long)n * 1024 + ty * 32) * 1024 + tx * 32;
        for (int i = tid; i < 32 * 32; i += 256) {
            int oyq = i / 32, oxq = i - oyq * 32;
            const float* p = &buf1[(2 * oyq) * STRIDE + 2 * oxq];
            float v = fmaxf(fmaxf(p[0], p[1]), fmaxf(p[STRIDE], p[STRIDE + 1]));
            out[outBase + (long)oyq * 1024 + oxq] = v;
        }
    }
}

extern "C" void kernel_launch(void* const* d_in, const int* in_sizes, int n_in,
                              void* d_out, int out_size, void* d_ws, size_t ws_size,
                              hipStream_t stream) {
    const float* x  = (const float*)d_in[0];
    const float* w1 = (const float*)d_in[1];
    const float* w2 = (const float*)d_in[2];
    const float* w3 = (const float*)d_in[3];
    const float* w4 = (const float*)d_in[4];
    float* out = (float*)d_out;

    // 4 images x (32x32) tiles of the 1024x1024 final output
    fused_conv_pool<<<dim3(4096), dim3(256), 0, stream>>>(x, w1, w2, w3, w4, out);
}